// GCN_layer_2834678415701
// MI455X (gfx1250) — compile-verified
//
#include <hip/hip_runtime.h>

// GCN layer for MI455X (gfx1250, wave32).
// Roofline: 154 MFLOP, ~50MB HBM traffic, ~1.2GB L2 traffic (gather + f32
// atomics). Feature matrix (19.2MB), output (19.2MB), deg/dis (200KB) all
// L2-resident. SpMM: 32 edges per wave, metadata loaded coalesced per-lane,
// broadcast via v_readlane -> scalar-base gathers + global_atomic_add_f32.

constexpr int D = 96; // feature dim (fixed by problem)

// --- zero workspace: deg[i]=0, present[i]=0 ---------------------------------
__global__ void k_init_ws(float* __restrict__ deg, int* __restrict__ present, int n) {
    int i = blockIdx.x * blockDim.x + threadIdx.x;
    if (i < n) { deg[i] = 0.0f; present[i] = 0; }
}

// --- present[index[i]] = 1 ---------------------------------------------------
__global__ void k_mark_present(const int* __restrict__ index,
                               int* __restrict__ present, int n) {
    int i = blockIdx.x * blockDim.x + threadIdx.x;
    if (i < n) present[index[i]] = 1;
}

// --- deg[row[e]] += val[e]  (800K f32 atomics into 200KB array -> L2) -------
__global__ void k_degree(const int* __restrict__ rows, const float* __restrict__ vals,
                         float* __restrict__ deg, int E) {
    int e = blockIdx.x * blockDim.x + threadIdx.x;
    if (e < E) unsafeAtomicAdd(&deg[rows[e]], vals[e]);
}

// --- deg -> d^{-1/2}, inf->0 (in place) -------------------------------------
__global__ void k_rsqrt(float* __restrict__ deg, int n) {
    int i = blockIdx.x * blockDim.x + threadIdx.x;
    if (i < n) {
        float d = deg[i];
        deg[i] = (d > 0.0f) ? rsqrtf(d) : 0.0f;
    }
}

// --- out[i] = present[row] ? 0 : features[i]  (float4 vectorized) -----------
__global__ void k_init_out(const float* __restrict__ feat, const int* __restrict__ present,
                           float* __restrict__ out, int n4) {
    int i = blockIdx.x * blockDim.x + threadIdx.x;
    if (i < n4) {
        int row = i / (D / 4);            // 24 float4 per row
        const float4* f4 = (const float4*)feat;
        float4* o4 = (float4*)out;
        float4 z = make_float4(0.f, 0.f, 0.f, 0.f);
        o4[i] = present[row] ? z : f4[i];
    }
}

// --- SpMM with fused permutation scatter, 32 edges per wave -----------------
// Lane l of wave w owns edge base+l: loads (row,col,val) coalesced, computes
// nv = val*dis[row]*dis[col] and dst = index[row]. Then the wave walks the 32
// edges; v_readlane broadcasts (col,dst,nv) into SGPRs so the 3 gather loads
// and 3 global_atomic_add_f32 per edge use scalar-base + imm-offset addressing.
__global__ void __launch_bounds__(256) k_spmm(
    const float* __restrict__ feat,
    const int*   __restrict__ rows,
    const int*   __restrict__ cols,
    const float* __restrict__ vals,
    const float* __restrict__ dis,
    const int*   __restrict__ index,
    float*       __restrict__ out,
    int E) {
    const int lane = threadIdx.x & 31;
    const int wave = blockIdx.x * (blockDim.x >> 5) + (threadIdx.x >> 5); // wave32
    const int base = wave * 32;
    if (base >= E) return;

    const int  e     = base + lane;
    const bool valid = e < E;
    const int  ec    = valid ? e : base;        // clamp tail lanes

    int   r   = rows[ec];
    int   c   = cols[ec];
    float v   = valid ? vals[ec] : 0.0f;
    float nv  = v * dis[r] * dis[c];
    int   dst = index[r];

    // Prefetch this batch's 32 source rows (3 x 128B lines each) into cache
    // before the broadcast loop consumes them -> global_prefetch_b8.
    const float* myrow = feat + (size_t)c * D;
#pragma unroll
    for (int k = 0; k < 3; ++k)
        __builtin_prefetch(myrow + 32 * k, 0, 0);

    const int cnt = (E - base < 32) ? (E - base) : 32;
    for (int i = 0; i < cnt; ++i) {
        const int   ci  = __builtin_amdgcn_readlane(c, i);
        const int   di  = __builtin_amdgcn_readlane(dst, i);
        const float nvi = __uint_as_float(
            __builtin_amdgcn_readlane((int)__float_as_uint(nv), i));

        const float* src = feat + (size_t)ci * D + lane;
        float*       dp  = out  + (size_t)di * D + lane;
#pragma unroll
        for (int k = 0; k < 3; ++k) {
            unsafeAtomicAdd(&dp[32 * k], src[32 * k] * nvi); // global_atomic_add_f32
        }
    }
}

extern "C" void kernel_launch(void* const* d_in, const int* in_sizes, int n_in,
                              void* d_out, int out_size, void* d_ws, size_t ws_size,
                              hipStream_t stream) {
    const float* feat  = (const float*)d_in[0];
    const int*   rows  = (const int*)  d_in[1];
    const int*   cols  = (const int*)  d_in[2];
    const float* vals  = (const float*)d_in[3];
    const int*   index = (const int*)  d_in[4];
    float*       out   = (float*)d_out;

    const int E = in_sizes[1];   // 800000
    const int N = in_sizes[4];   // 50000

    // workspace layout: [0,N) f32 deg (becomes d^-1/2 in place), [N,2N) i32 present
    float* deg     = (float*)d_ws;
    int*   present = (int*)d_ws + N;

    const int T = 256;
    k_init_ws     <<<(N + T - 1) / T, T, 0, stream>>>(deg, present, N);
    k_mark_present<<<(N + T - 1) / T, T, 0, stream>>>(index, present, N);
    k_degree      <<<(E + T - 1) / T, T, 0, stream>>>(rows, vals, deg, E);
    k_rsqrt       <<<(N + T - 1) / T, T, 0, stream>>>(deg, N);

    const int n4 = N * (D / 4);
    k_init_out    <<<(n4 + T - 1) / T, T, 0, stream>>>(feat, present, out, n4);

    // 32 edges per wave, 8 waves per block -> 256 edges per block
    const int edgesPerBlock = (T / 32) * 32;  // 256
    const int nBlocks = (E + edgesPerBlock - 1) / edgesPerBlock;
    k_spmm<<<nBlocks, T, 0, stream>>>(feat, rows, cols, vals, deg, index, out, E);
}